// NET_Wrapper_20418274525324
// MI455X (gfx1250) — compile-verified
//
#include <hip/hip_runtime.h>
#include <math.h>

// ---------------------------------------------------------------------------
// Dimensions (fixed by the reference)
// ---------------------------------------------------------------------------
#define TDIM 300
#define NFQ  161
#define NC   5
#define HID  1024
#define G4   4096   // 4*HID

typedef float v2f __attribute__((ext_vector_type(2)));
typedef float v8f __attribute__((ext_vector_type(8)));

static __device__ __forceinline__ v8f wmma_f32(v2f a, v2f b, v8f c) {
  // D = A(16x4,f32) * B(4x16,f32) + C(16x16,f32)
  return __builtin_amdgcn_wmma_f32_16x16x4_f32(false, a, false, b, (short)0, c,
                                               false, false);
}

static __device__ __forceinline__ v8f zero8() {
  v8f z = {0.f, 0.f, 0.f, 0.f, 0.f, 0.f, 0.f, 0.f};
  return z;
}

// ---------------------------------------------------------------------------
// Workspace layout (float offsets). Total ~26.72M floats (~107 MB).
// Early region (x0/enc/lstm/pre/cat/tmp) is recycled for covariances after
// the decoders finish.
// ---------------------------------------------------------------------------
static const size_t OFF_X0      = 0;          // (B,10,T,161)   966000
static const size_t OFF_E1      = 966000;     // (B,16,T,80)    768000
static const size_t OFF_E2      = 1734000;    // (B,32,T,39)    748800
static const size_t OFF_E3      = 2482800;    // (B,64,T,19)    729600
static const size_t OFF_E4      = 3212400;    // (B,128,T,9)    691200
static const size_t OFF_E5      = 3903600;    // (B,256,T,4)    614400
static const size_t OFF_Z0      = 4518000;    // (B,T,1024)     614400
static const size_t OFF_Z1      = 5132400;
static const size_t OFF_Z2      = 5746800;
static const size_t OFF_LSTMOUT = 6361200;    // (B,256,T,4)    614400
static const size_t OFF_PRE     = 6975600;    // (B,T,4096)    2457600
static const size_t OFF_CAT     = 9433200;    //               1536000
static const size_t OFF_TMP     = 10969200;   //               1536000
static const size_t OFF_MASK    = 12505200;   // (B,1,T,161)     96600
static const size_t OFF_Q1      = 12601800;   // (B,24,T,161)  2318400
static const size_t OFF_K1      = 14920200;
static const size_t OFF_Q2      = 17238600;
static const size_t OFF_K2      = 19557000;
static const size_t OFF_PHIS    = 21875400;   // (B,F,T,50)    4830000
// ---- reuse of dead early region (after decoders) ----
static const size_t OFF_COVS    = 0;          // (B,F,T,50)    4830000
static const size_t OFF_COVN    = 4830000;    //               4830000
static const size_t OFF_MEANS   = 9660000;    // (B,T,5,2)        6000
static const size_t OFF_MEANN   = 9666000;    //                  6000
// ---- LSTM scratch (after PHIS) ----
static const size_t OFF_HBUF    = 26705400;   // (2,1024)
static const size_t OFF_CBUF    = 26707448;
static const size_t OFF_ZBUF    = 26709496;   // (2,4096)
static const size_t OFF_CNT     = 26717688;   // barrier counter (int)

// ---------------------------------------------------------------------------
// Input layout transform: x0[b, 2c+ri, t, f] = spec[b,c,t,f,ri]
// ---------------------------------------------------------------------------
__global__ void k_in_tr(const float* __restrict__ SPEC, float* __restrict__ X0) {
  int idx = blockIdx.x * blockDim.x + threadIdx.x;
  int total = 2 * 10 * TDIM * NFQ;
  if (idx >= total) return;
  int f = idx % NFQ; int r = idx / NFQ;
  int t = r % TDIM;  r /= TDIM;
  int ch = r % 10;   int b = r / 10;
  int c = ch >> 1, ri = ch & 1;
  X0[idx] = SPEC[(((size_t)(b * NC + c) * TDIM + t) * NFQ + f) * 2 + ri];
}

// ---------------------------------------------------------------------------
// Encoder conv block: kernel (2,3), stride (1,2), time pad (1,0), ELU, BN
// ---------------------------------------------------------------------------
__global__ void k_conv(const float* __restrict__ X, int Ci, int Fin,
                       const float* __restrict__ Wt, const float* __restrict__ bias,
                       const float* __restrict__ gam, const float* __restrict__ bet,
                       const float* __restrict__ mu, const float* __restrict__ var,
                       float* __restrict__ Y, int Co, int Fout) {
  int idx = blockIdx.x * blockDim.x + threadIdx.x;
  int total = 2 * Co * TDIM * Fout;
  if (idx >= total) return;
  int fo = idx % Fout; int r = idx / Fout;
  int t = r % TDIM;    r /= TDIM;
  int co = r % Co;     int b = r / Co;
  float acc = bias[co];
  for (int ci = 0; ci < Ci; ++ci) {
    const float* wr = Wt + (size_t)(co * Ci + ci) * 6;
    const float* xr = X + ((size_t)(b * Ci + ci) * TDIM) * Fin + 2 * fo;
#pragma unroll
    for (int kt = 0; kt < 2; ++kt) {
      int tin = t + kt - 1;
      if (tin < 0) continue;
      const float* xx = xr + (size_t)tin * Fin;
      const float* ww = wr + kt * 3;
      acc += ww[0] * xx[0] + ww[1] * xx[1] + ww[2] * xx[2];
    }
  }
  float e = acc > 0.f ? acc : expm1f(acc);
  Y[idx] = (e - mu[co]) * rsqrtf(var[co] + 1e-5f) * gam[co] + bet[co];
}

// ---------------------------------------------------------------------------
// Transposed conv block (freq stride 2, pad(2,2+op)) + time front pad + ELU+BN
// y[t,fo] (t>=1): sum_ci sum_kt sum_kf w[ci,co,1-kt,2-kf]*x[ci,t-1+kt,(fo+kf-2)/2]
// ---------------------------------------------------------------------------
__global__ void k_deconv(const float* __restrict__ X, int Ci, int Fin,
                         const float* __restrict__ Wt, const float* __restrict__ bias,
                         const float* __restrict__ gam, const float* __restrict__ bet,
                         const float* __restrict__ mu, const float* __restrict__ var,
                         float* __restrict__ Y, int Co, int Fout) {
  int idx = blockIdx.x * blockDim.x + threadIdx.x;
  int total = 2 * Co * TDIM * Fout;
  if (idx >= total) return;
  int fo = idx % Fout; int r = idx / Fout;
  int t = r % TDIM;    r /= TDIM;
  int co = r % Co;     int b = r / Co;
  float acc = 0.f;
  if (t > 0) {
    acc = bias[co];
    int tp = t - 1;
    for (int ci = 0; ci < Ci; ++ci) {
      const float* xrow = X + ((size_t)(b * Ci + ci) * TDIM + tp) * Fin;
      const float* wrow = Wt + (size_t)(ci * Co + co) * 6;
#pragma unroll
      for (int kt = 0; kt < 2; ++kt) {
        const float* xr = xrow + (size_t)kt * Fin;
        const float* wr = wrow + (1 - kt) * 3;
#pragma unroll
        for (int kf = 0; kf < 3; ++kf) {
          int g = fo + kf - 2;
          if (g >= 0 && !(g & 1)) {
            int fi = g >> 1;
            if (fi < Fin) acc += wr[2 - kf] * xr[fi];
          }
        }
      }
    }
  }
  float e = acc > 0.f ? acc : expm1f(acc);
  Y[idx] = (e - mu[co]) * rsqrtf(var[co] + 1e-5f) * gam[co] + bet[co];
}

// ---------------------------------------------------------------------------
// Channel concat: dst(B,Ca+Cb,T,F) = [A | B]
// ---------------------------------------------------------------------------
__global__ void k_concat(float* __restrict__ D, const float* __restrict__ A, int Ca,
                         const float* __restrict__ Bs, int Cb, int F) {
  int idx = blockIdx.x * blockDim.x + threadIdx.x;
  int Ct = Ca + Cb;
  int total = 2 * Ct * TDIM * F;
  if (idx >= total) return;
  int f = idx % F; int r = idx / F;
  int t = r % TDIM; r /= TDIM;
  int c = r % Ct;   int b = r / Ct;
  float v;
  if (c < Ca) v = A[((size_t)(b * Ca + c) * TDIM + t) * F + f];
  else        v = Bs[((size_t)(b * Cb + (c - Ca)) * TDIM + t) * F + f];
  D[idx] = v;
}

// ---------------------------------------------------------------------------
// (B,256,T,4) -> (B,T,1024) and back
// ---------------------------------------------------------------------------
__global__ void k_pack_z0(const float* __restrict__ E5, float* __restrict__ Z0) {
  int idx = blockIdx.x * blockDim.x + threadIdx.x;
  int total = 2 * TDIM * HID;
  if (idx >= total) return;
  int d = idx % HID; int r = idx / HID;
  int t = r % TDIM;  int b = r / TDIM;
  int ch = d >> 2, fq = d & 3;
  Z0[idx] = E5[((size_t)(b * 256 + ch) * TDIM + t) * 4 + fq];
}

__global__ void k_pack_lstmout(const float* __restrict__ Z2, float* __restrict__ LO) {
  int idx = blockIdx.x * blockDim.x + threadIdx.x;
  int total = 2 * 256 * TDIM * 4;
  if (idx >= total) return;
  int fq = idx & 3; int r = idx >> 2;
  int t = r % TDIM; r /= TDIM;
  int ch = r % 256; int b = r / 256;
  LO[idx] = Z2[((size_t)(b * TDIM + t)) * HID + ch * 4 + fq];
}

// ---------------------------------------------------------------------------
// LSTM input projection: pre[m, g] = X[m,:]·Wi[g,:] + bi[g] + bh[g]
// M = 600, N = 4096, K = 1024. One wave per 16x16 tile, fp32 WMMA.
// A loads are unconditional from a clamped row: rows >= M produce garbage
// only in D rows >= M, which are masked at store time.
// ---------------------------------------------------------------------------
__global__ __launch_bounds__(32)
void k_gemm_pre(const float* __restrict__ X, const float* __restrict__ Wi,
                const float* __restrict__ bi, const float* __restrict__ bh,
                float* __restrict__ Y, int M) {
  int mt = blockIdx.x, nt = blockIdx.y;
  int lane = threadIdx.x;
  int half = lane >> 4, l16 = lane & 15;
  int mrow = mt * 16 + l16;
  int ncol = nt * 16 + l16;
  int mclamp = mrow < M ? mrow : M - 1;
  const float* xrow = X + (size_t)mclamp * HID;
  const float* wrow = Wi + (size_t)ncol * HID;
  v8f acc = zero8();
#pragma unroll 4
  for (int k0 = 0; k0 < HID; k0 += 4) {
    int ka = k0 + half * 2;
    float2 t2 = *(const float2*)(xrow + ka);
    float2 w2 = *(const float2*)(wrow + ka);
    v2f a, b;
    a.x = t2.x; a.y = t2.y;
    b.x = w2.x; b.y = w2.y;
    acc = wmma_f32(a, b, acc);
  }
  float bsum = bi[ncol] + bh[ncol];
#pragma unroll
  for (int r = 0; r < 8; ++r) {
    int row = mt * 16 + r + half * 8;
    if (row < M) Y[(size_t)row * G4 + ncol] = acc[r] + bsum;
  }
}

// ---------------------------------------------------------------------------
// Persistent LSTM recurrence. 32 blocks x 256 threads (256 waves).
// Each wave owns one 16-column tile of the 4096 gate pre-activations.
// Global monotonic phase barrier (2 phases per timestep).
// ---------------------------------------------------------------------------
__device__ __forceinline__ void gbar(int* cnt, int nb, int phase) {
  __syncthreads();
  if (threadIdx.x == 0) {
    __threadfence();
    atomicAdd(cnt, 1);
    while (atomicAdd(cnt, 0) < nb * phase) { __builtin_amdgcn_s_sleep(1); }
    __threadfence();
  }
  __syncthreads();
}

__global__ void k_lstm_init(float* h, float* c, int* cnt) {
  int i = blockIdx.x * blockDim.x + threadIdx.x;
  if (i < 2 * HID) { h[i] = 0.f; c[i] = 0.f; }
  if (i == 0) *cnt = 0;
}

__global__ __launch_bounds__(256)
void k_lstm_seq(const float* __restrict__ pre,  // (B,T,4096)
                const float* __restrict__ Wh,   // (4096,1024)
                float* __restrict__ zout,       // (B,T,1024)
                float* hbuf, float* cbuf, float* zbuf, int* cnt) {
  int tid = threadIdx.x;
  int wid = blockIdx.x * (blockDim.x >> 5) + (tid >> 5);
  int lane = tid & 31;
  int half = lane >> 4, l16 = lane & 15;
  int ncol = wid * 16 + l16;
  const float* wrow = Wh + (size_t)ncol * HID;
  // A rows >= 2 are unused in D; load from a valid clamped address instead of
  // branching (avoids EXEC-mask churn in the hot loop).
  const float* hrow = hbuf + (size_t)(l16 & 1) * HID;
  int nb = gridDim.x;
  for (int t = 0; t < TDIM; ++t) {
    v8f acc = zero8();
#pragma unroll 4
    for (int k0 = 0; k0 < HID; k0 += 4) {
      int ka = k0 + half * 2;
      float2 h2 = *(const float2*)(hrow + ka);
      float2 w2 = *(const float2*)(wrow + ka);
      v2f a, b;
      a.x = h2.x; a.y = h2.y;
      b.x = w2.x; b.y = w2.y;
      acc = wmma_f32(a, b, acc);
    }
    if (half == 0) {  // rows 0,1 hold batch 0,1
      zbuf[0 * G4 + ncol] = acc[0] + pre[((size_t)(0 * TDIM + t)) * G4 + ncol];
      zbuf[1 * G4 + ncol] = acc[1] + pre[((size_t)(1 * TDIM + t)) * G4 + ncol];
    }
    gbar(cnt, nb, 2 * t + 1);
    int gidx = blockIdx.x * blockDim.x + tid;
    if (gidx < 2 * HID) {
      int b = gidx >> 10, u = gidx & (HID - 1);
      float zi = zbuf[b * G4 + u];
      float zf = zbuf[b * G4 + HID + u];
      float zg = zbuf[b * G4 + 2 * HID + u];
      float zo = zbuf[b * G4 + 3 * HID + u];
      float ii = 1.f / (1.f + expf(-zi));
      float ff = 1.f / (1.f + expf(-zf));
      float gg = tanhf(zg);
      float oo = 1.f / (1.f + expf(-zo));
      float c = ff * cbuf[gidx] + ii * gg;
      float h = oo * tanhf(c);
      cbuf[gidx] = c; hbuf[gidx] = h;
      zout[((size_t)(b * TDIM + t)) * HID + u] = h;
    }
    gbar(cnt, nb, 2 * t + 2);
  }
}

// ---------------------------------------------------------------------------
// Covariance: per-frequency-centered complex outer products
// ---------------------------------------------------------------------------
__global__ void k_covmean(const float* __restrict__ SPEC, const float* __restrict__ MASK,
                          float* __restrict__ MS, float* __restrict__ MN) {
  int idx = blockIdx.x * blockDim.x + threadIdx.x;
  if (idx >= 2 * TDIM * NC) return;
  int c = idx % NC; int r = idx / NC;
  int t = r % TDIM; int b = r / TDIM;
  float sr = 0, si = 0, nr = 0, ni = 0;
  const float* sp = SPEC + (((size_t)(b * NC + c) * TDIM + t) * NFQ) * 2;
  const float* mk = MASK + ((size_t)b * TDIM + t) * NFQ;
  for (int f = 0; f < NFQ; ++f) {
    float re = sp[2 * f], im = sp[2 * f + 1], m = mk[f];
    sr += m * re; si += m * im;
    nr += (1.f - m) * re; ni += (1.f - m) * im;
  }
  float invF = 1.f / (float)NFQ;
  MS[idx * 2] = sr * invF; MS[idx * 2 + 1] = si * invF;
  MN[idx * 2] = nr * invF; MN[idx * 2 + 1] = ni * invF;
}

__global__ void k_cov(const float* __restrict__ SPEC, const float* __restrict__ MASK,
                      const float* __restrict__ MS, const float* __restrict__ MN,
                      float* __restrict__ CS, float* __restrict__ CN) {
  int idx = blockIdx.x * blockDim.x + threadIdx.x;
  if (idx >= 2 * NFQ * TDIM) return;
  int t = idx % TDIM; int r = idx / TDIM;
  int f = r % NFQ;    int b = r / NFQ;
  float m = MASK[((size_t)b * TDIM + t) * NFQ + f];
  float zsr[NC], zsi[NC], znr[NC], zni[NC];
#pragma unroll
  for (int c = 0; c < NC; ++c) {
    size_t si = (((size_t)(b * NC + c) * TDIM + t) * NFQ + f) * 2;
    float re = SPEC[si], im = SPEC[si + 1];
    size_t mi = ((size_t)(b * TDIM + t) * NC + c) * 2;
    zsr[c] = m * re - MS[mi];         zsi[c] = m * im - MS[mi + 1];
    znr[c] = (1.f - m) * re - MN[mi]; zni[c] = (1.f - m) * im - MN[mi + 1];
  }
  float* cs = CS + (size_t)idx * 50;
  float* cn = CN + (size_t)idx * 50;
  const float inv = 1.f / (float)(NFQ - 1);
#pragma unroll
  for (int c1 = 0; c1 < NC; ++c1)
#pragma unroll
    for (int c2 = 0; c2 < NC; ++c2) {
      int kk = (c1 * NC + c2) * 2;
      cs[kk]     = (zsr[c1] * zsr[c2] + zsi[c1] * zsi[c2]) * inv;
      cs[kk + 1] = (zsi[c1] * zsr[c2] - zsr[c1] * zsi[c2]) * inv;
      cn[kk]     = (znr[c1] * znr[c2] + zni[c1] * zni[c2]) * inv;
      cn[kk + 1] = (zni[c1] * znr[c2] - znr[c1] * zni[c2]) * inv;
    }
}

// ---------------------------------------------------------------------------
// Complex helpers for the MVDR solve
// ---------------------------------------------------------------------------
struct cpx { float re, im; };
static __device__ __forceinline__ cpx cmul(cpx a, cpx b) {
  cpx r; r.re = a.re * b.re - a.im * b.im; r.im = a.re * b.im + a.im * b.re; return r;
}
static __device__ __forceinline__ cpx csub(cpx a, cpx b) {
  cpx r; r.re = a.re - b.re; r.im = a.im - b.im; return r;
}
static __device__ __forceinline__ cpx crecip(cpx a) {
  float d = a.re * a.re + a.im * a.im;
  float inv = 1.f / d;
  cpx r; r.re = a.re * inv; r.im = -a.im * inv; return r;
}

// ---------------------------------------------------------------------------
// Fused causal attention (flash-style) + optional MVDR solve.
// One block per (b,f). mode 0: write Phi. mode 1: read PhiS, solve, write OUT.
// S = softmax(mask(QK^T/sqrt(24))) ; Phi = S @ V (re & im separately)
// ---------------------------------------------------------------------------
__global__ __launch_bounds__(256)
void k_attn(const float* __restrict__ Q, const float* __restrict__ K,  // (B,24,T,F)
            const float* __restrict__ COV,                              // (B,F,T,50)
            float* __restrict__ PHI, const float* __restrict__ PHIS,
            const float* __restrict__ SPEC, float* __restrict__ OUT,
            int mode) {
  __shared__ float S[16][308];
  __shared__ float ACC[16][52];
  const int T = TDIM, F = NFQ;
  int bf = blockIdx.x;
  int b = bf / F, f = bf % F;
  int tid = threadIdx.x;
  int wave = tid >> 5, lane = tid & 31;
  int half = lane >> 4, l16 = lane & 15;
  const float scale = rsqrtf(24.f);
  const float* covbase = COV + (size_t)bf * T * 50;
  const int nM = (T + 15) / 16;  // 19

  for (int mt = 0; mt < nM; ++mt) {
    int trow = mt * 16 + l16;
    int trA = trow < T ? trow : T - 1;
    // ---- S tile = Q K^T --------------------------------------------------
    for (int nt = wave; nt < nM; nt += 8) {
      int scol = nt * 16 + l16;
      int scB = scol < T ? scol : T - 1;
      v8f acc = zero8();
#pragma unroll
      for (int k0 = 0; k0 < 24; k0 += 4) {
        int ka = k0 + half * 2;
        v2f a, bb;
        a.x  = Q[((size_t)(b * 24 + ka)     * T + trA) * F + f];
        a.y  = Q[((size_t)(b * 24 + ka + 1) * T + trA) * F + f];
        bb.x = K[((size_t)(b * 24 + ka)     * T + scB) * F + f];
        bb.y = K[((size_t)(b * 24 + ka + 1) * T + scB) * F + f];
        acc = wmma_f32(a, bb, acc);
      }
#pragma unroll
      for (int r = 0; r < 8; ++r) S[r + half * 8][nt * 16 + l16] = acc[r];
    }
    __syncthreads();
    // ---- causal mask + softmax (16 threads per row) ----------------------
    {
      int row = tid >> 4;
      int j = tid & 15;
      int trm = mt * 16 + row;
      int ncols = nM * 16;
      float mx = -3.0e38f;
      for (int c = j; c < ncols; c += 16) {
        float v = (trm < T && c < T && c <= trm) ? S[row][c] * scale : -1.0e30f;
        S[row][c] = v;
        mx = fmaxf(mx, v);
      }
#pragma unroll
      for (int off = 1; off < 16; off <<= 1) mx = fmaxf(mx, __shfl_xor(mx, off, 32));
      float sum = 0.f;
      for (int c = j; c < ncols; c += 16) {
        float e = expf(S[row][c] - mx);
        S[row][c] = e;
        sum += e;
      }
#pragma unroll
      for (int off = 1; off < 16; off <<= 1) sum += __shfl_xor(sum, off, 32);
      float invs = 1.f / sum;
      for (int c = j; c < ncols; c += 16) S[row][c] *= invs;
    }
    __syncthreads();
    for (int i = tid; i < 16 * 52; i += 256) ((float*)ACC)[i] = 0.f;
    __syncthreads();
    // ---- Phi tile = P @ V  (re/im, N=25 in two 16-tiles, K split 2-way) --
    {
      int ntile = wave & 3;       // 0: re[0..15] 1: re[16..24] 2: im[0..15] 3: im[16..24]
      int ks = wave >> 2;         // K split half
      int reim = ntile >> 1;
      int n = (ntile & 1) * 16 + l16;
      bool nval = n < 25;
      int nc = nval ? n : 24;     // clamped, always-valid cov column
      int klo = ks ? 148 : 0;
      int khi = ks ? 300 : 148;
      v8f acc = zero8();
      for (int k0 = klo; k0 < khi; k0 += 4) {
        int ka = k0 + half * 2;
        v2f a, bb;
        a.x = S[l16][ka];
        a.y = S[l16][ka + 1];
        bb.x = covbase[(size_t)ka       * 50 + 2 * nc + reim];
        bb.y = covbase[(size_t)(ka + 1) * 50 + 2 * nc + reim];
        acc = wmma_f32(a, bb, acc);
      }
      if (nval) {
#pragma unroll
        for (int r = 0; r < 8; ++r)
          atomicAdd(&ACC[r + half * 8][2 * n + reim], acc[r]);
      }
    }
    __syncthreads();
    if (mode == 0) {
      for (int i = tid; i < 16 * 50; i += 256) {
        int row = i / 50, cc = i % 50;
        int trm = mt * 16 + row;
        if (trm < T) PHI[((size_t)bf * T + trm) * 50 + cc] = ACC[row][cc];
      }
    } else if (tid < 16) {
      int trm = mt * 16 + tid;
      if (trm < T) {
        cpx A[5][5], Bm[5][5];
        const float* ps = PHIS + ((size_t)bf * T + trm) * 50;
#pragma unroll
        for (int i = 0; i < 5; ++i)
#pragma unroll
          for (int j = 0; j < 5; ++j) {
            int kk = (i * 5 + j) * 2;
            A[i][j].re = ACC[tid][kk];  A[i][j].im = ACC[tid][kk + 1];
            Bm[i][j].re = ps[kk];       Bm[i][j].im = ps[kk + 1];
          }
#pragma unroll
        for (int i = 0; i < 5; ++i) { A[i][i].re += 1e-7f; A[i][i].im += 1e-7f; }
        // Gauss-Jordan with partial pivoting: Bm <- A^{-1} Bm
        for (int col = 0; col < 5; ++col) {
          int p = col;
          float best = A[col][col].re * A[col][col].re + A[col][col].im * A[col][col].im;
          for (int rr = col + 1; rr < 5; ++rr) {
            float m2 = A[rr][col].re * A[rr][col].re + A[rr][col].im * A[rr][col].im;
            if (m2 > best) { best = m2; p = rr; }
          }
          if (p != col) {
            for (int j = 0; j < 5; ++j) {
              cpx tA = A[col][j]; A[col][j] = A[p][j]; A[p][j] = tA;
              cpx tB = Bm[col][j]; Bm[col][j] = Bm[p][j]; Bm[p][j] = tB;
            }
          }
          cpx inv = crecip(A[col][col]);
          for (int j = 0; j < 5; ++j) {
            A[col][j] = cmul(A[col][j], inv);
            Bm[col][j] = cmul(Bm[col][j], inv);
          }
          for (int rr = 0; rr < 5; ++rr) {
            if (rr == col) continue;
            cpx fac = A[rr][col];
            for (int j = 0; j < 5; ++j) {
              A[rr][j] = csub(A[rr][j], cmul(fac, A[col][j]));
              Bm[rr][j] = csub(Bm[rr][j], cmul(fac, Bm[col][j]));
            }
          }
        }
        cpx tr; tr.re = 0.f; tr.im = 0.f;
#pragma unroll
        for (int i = 0; i < 5; ++i) { tr.re += Bm[i][i].re; tr.im += Bm[i][i].im; }
        tr.re += 1.1920929e-7f; tr.im += 1.1920929e-7f;
        cpx invtr = crecip(tr);
        float Sre = 0.f, Sim = 0.f;
#pragma unroll
        for (int c = 0; c < 5; ++c) {
          cpx Wc = cmul(Bm[c][0], invtr);
          size_t si = (((size_t)(b * NC + c) * T + trm) * F + f) * 2;
          float re = SPEC[si], im = SPEC[si + 1];
          Sre += Wc.re * re + Wc.im * im;   // conj(W)*stft
          Sim += Wc.re * im - Wc.im * re;
        }
        OUT[((size_t)(b * T + trm) * F + f) * 2]     = Sre;
        OUT[((size_t)(b * T + trm) * F + f) * 2 + 1] = Sim;
      }
    }
    __syncthreads();
  }
}

// ---------------------------------------------------------------------------
// Host orchestration
// ---------------------------------------------------------------------------
extern "C" void kernel_launch(void* const* d_in, const int* in_sizes, int n_in,
                              void* d_out, int out_size, void* d_ws, size_t ws_size,
                              hipStream_t stream) {
  (void)in_sizes; (void)n_in; (void)out_size; (void)ws_size;
  const float* SPEC = (const float*)d_in[0];
  float* ws = (float*)d_ws;
  auto P = [&](int i) { return (const float*)d_in[i]; };
  auto nblk = [](long n) { return (unsigned)((n + 255) / 256); };

  float* x0   = ws + OFF_X0;
  float* enc[5] = { ws + OFF_E1, ws + OFF_E2, ws + OFF_E3, ws + OFF_E4, ws + OFF_E5 };
  float* z0   = ws + OFF_Z0;
  float* z1   = ws + OFF_Z1;
  float* z2   = ws + OFF_Z2;
  float* lo   = ws + OFF_LSTMOUT;
  float* pre  = ws + OFF_PRE;
  float* cat  = ws + OFF_CAT;
  float* tmp  = ws + OFF_TMP;
  float* maskb = ws + OFF_MASK;
  float* qk[4] = { ws + OFF_Q1, ws + OFF_K1, ws + OFF_Q2, ws + OFF_K2 };
  float* phis = ws + OFF_PHIS;
  float* covs = ws + OFF_COVS;
  float* covn = ws + OFF_COVN;
  float* means = ws + OFF_MEANS;
  float* meann = ws + OFF_MEANN;
  float* hbuf = ws + OFF_HBUF;
  float* cbuf = ws + OFF_CBUF;
  float* zbuf = ws + OFF_ZBUF;
  int*   cnt  = (int*)(ws + OFF_CNT);

  // 1) input transpose
  k_in_tr<<<nblk(2L * 10 * TDIM * NFQ), 256, 0, stream>>>(SPEC, x0);

  // 2) encoder
  const int ECi[5] = { 10, 16, 32, 64, 128 };
  const int ECo[5] = { 16, 32, 64, 128, 256 };
  const int EF[6]  = { 161, 80, 39, 19, 9, 4 };
  const float* ein = x0;
  for (int l = 0; l < 5; ++l) {
    int base = 1 + l * 6;
    long n = 2L * ECo[l] * TDIM * EF[l + 1];
    k_conv<<<nblk(n), 256, 0, stream>>>(ein, ECi[l], EF[l],
        P(base), P(base + 1), P(base + 2), P(base + 3), P(base + 4), P(base + 5),
        enc[l], ECo[l], EF[l + 1]);
    ein = enc[l];
  }

  // 3) LSTM (2 layers)
  k_pack_z0<<<nblk(2L * TDIM * HID), 256, 0, stream>>>(enc[4], z0);
  const float* lin[2] = { z0, z1 };
  float* lout[2] = { z1, z2 };
  for (int l = 0; l < 2; ++l) {
    int base = 31 + l * 4;  // Wi, Wh, bi, bh
    dim3 gg((600 + 15) / 16, G4 / 16);
    k_gemm_pre<<<gg, 32, 0, stream>>>(lin[l], P(base), P(base + 2), P(base + 3),
                                      pre, 600);
    k_lstm_init<<<8, 256, 0, stream>>>(hbuf, cbuf, cnt);
    k_lstm_seq<<<32, 256, 0, stream>>>(pre, P(base + 1), lout[l], hbuf, cbuf,
                                       zbuf, cnt);
  }
  k_pack_lstmout<<<nblk(2L * 256 * TDIM * 4), 256, 0, stream>>>(z2, lo);

  // 4) decoders (mask, q1, k1, q2, k2)
  const int DCo[4] = { 128, 64, 32, 16 };
  const int DF[6]  = { 4, 9, 19, 39, 80, 161 };
  const int SKC[5] = { 256, 128, 64, 32, 16 };
  float* skips[5] = { enc[4], enc[3], enc[2], enc[1], enc[0] };
  for (int br = 0; br < 5; ++br) {
    int last = (br == 0) ? 1 : 24;
    float* fin = (br == 0) ? maskb : qk[br - 1];
    float* h = lo;
    int hc = 256;
    for (int l = 0; l < 5; ++l) {
      int Ci = hc + SKC[l];
      int Co = (l == 4) ? last : DCo[l];
      int Fin = DF[l], Fout = DF[l + 1];
      int base = 39 + br * 30 + l * 6;
      k_concat<<<nblk(2L * Ci * TDIM * Fin), 256, 0, stream>>>(cat, h, hc,
                                                               skips[l], SKC[l], Fin);
      float* dst = (l == 4) ? fin : tmp;
      k_deconv<<<nblk(2L * Co * TDIM * Fout), 256, 0, stream>>>(cat, Ci, Fin,
          P(base), P(base + 1), P(base + 2), P(base + 3), P(base + 4), P(base + 5),
          dst, Co, Fout);
      h = dst; hc = Co;
    }
  }

  // 5) covariances (reuses early workspace region)
  k_covmean<<<nblk(2L * TDIM * NC), 256, 0, stream>>>(SPEC, maskb, means, meann);
  k_cov<<<nblk(2L * NFQ * TDIM), 256, 0, stream>>>(SPEC, maskb, means, meann,
                                                   covs, covn);

  // 6) attention pass 1: Phi_S  (q1, k1, cov_s)
  k_attn<<<2 * NFQ, 256, 0, stream>>>(qk[0], qk[1], covs, phis, nullptr, nullptr,
                                      nullptr, 0);
  // 7) attention pass 2 + fused MVDR solve: (q2, k2, cov_n) -> output
  k_attn<<<2 * NFQ, 256, 0, stream>>>(qk[2], qk[3], covn, nullptr, phis, SPEC,
                                      (float*)d_out, 1);
}